// GCNDHLA_10153302687981
// MI455X (gfx1250) — compile-verified
//
#include <hip/hip_runtime.h>

#define NN 100000
#define NE 1600000
#define NH 128      // N_FEAT == N_HID == 128
#define NC 16

typedef float v2f __attribute__((ext_vector_type(2)));
typedef float v8f __attribute__((ext_vector_type(8)));

// ---------------- degree / normalization ----------------
__global__ __launch_bounds__(256) void k_init_cnt(unsigned int* cnt, int n) {
  int i = blockIdx.x * blockDim.x + threadIdx.x;
  if (i < n) cnt[i] = 1u;                      // self-loop contributes 1
}

__global__ __launch_bounds__(256) void k_count(const int* __restrict__ row,
                                               unsigned int* cnt, int e) {
  int i = blockIdx.x * blockDim.x + threadIdx.x;
  if (i < e) atomicAdd(&cnt[row[i]], 1u);
}

__global__ __launch_bounds__(256) void k_dinv(float* dinv, int n) {
  int i = blockIdx.x * blockDim.x + threadIdx.x;
  if (i < n) {
    unsigned int c = ((const unsigned int*)dinv)[i];
    dinv[i] = rsqrtf((float)c);               // deg >= 1 always
  }
}

// ---------------- GEMM: C[M,128] = A[M,128] * B[128,128], WMMA f32 16x16x4 ----
// One block = 16 output rows; A tile staged in LDS; 8 waves x 16 cols.
__global__ __launch_bounds__(256) void k_gemm128(const float* __restrict__ A,
                                                 const float* __restrict__ B,
                                                 float* __restrict__ C, int M) {
  __shared__ float tA[16 * NH];
  const int tid = threadIdx.x;
  const int m0 = blockIdx.x * 16;
  #pragma unroll
  for (int j = tid; j < 16 * NH; j += 256) {
    int r = j >> 7, c = j & 127;
    tA[j] = A[(size_t)(m0 + r) * NH + c];
  }
  __syncthreads();

  const int wave = tid >> 5;
  const int lane = tid & 31;
  const int half = lane >> 4;       // 0: K even pair base, 1: +2
  const int l15  = lane & 15;
  const int n0   = wave * 16;

  v8f acc = {};
  const float* arow = tA + l15 * NH;          // A row for this lane (M = l15)
  #pragma unroll 4
  for (int k0 = 0; k0 < NH; k0 += 4) {
    int k = k0 + 2 * half;
    v2f a = { arow[k], arow[k + 1] };                             // A 16x4 layout
    v2f b = { B[k * NH + n0 + l15], B[(k + 1) * NH + n0 + l15] }; // B 4x16 layout
    acc = __builtin_amdgcn_wmma_f32_16x16x4_f32(
        false, a, false, b, (short)0, acc, false, false);
  }

  const int colc  = n0 + l15;
  const int rbase = m0 + 8 * half;            // C/D: VGPR r -> row m0+r (+8 hi half)
  #pragma unroll
  for (int r = 0; r < 8; ++r)
    C[(size_t)(rbase + r) * NH + colc] = acc[r];
}

// ---------------- self-loop init: agg[i] = hw[i] * dinv[node]^2 --------------
__global__ __launch_bounds__(256) void k_selfloop(float* __restrict__ agg,
                                                  const float* __restrict__ hw,
                                                  const float* __restrict__ dinv,
                                                  int total) {
  int i = blockIdx.x * blockDim.x + threadIdx.x;
  if (i < total) {
    float d = dinv[i >> 7];
    agg[i] = hw[i] * d * d;
  }
}

// ---------------- edge scatter: agg[row] += hw[col] * ew ---------------------
// One wave per edge, float4 per lane, hardware fp32 atomics (L2-resident).
__global__ __launch_bounds__(256) void k_scatter(const int* __restrict__ row,
                                                 const int* __restrict__ col,
                                                 const float* __restrict__ dinv,
                                                 const float* __restrict__ hw,
                                                 float* __restrict__ agg, int e) {
  int ei = blockIdx.x * 8 + (threadIdx.x >> 5);
  if (ei >= e) return;
  int lane = threadIdx.x & 31;
  int r = row[ei], c = col[ei];
  float w = dinv[r] * dinv[c];
  float4 v = ((const float4*)(hw + (size_t)c * NH))[lane];
  float* dst = agg + (size_t)r * NH + lane * 4;
  unsafeAtomicAdd(dst + 0, v.x * w);
  unsafeAtomicAdd(dst + 1, v.y * w);
  unsafeAtomicAdd(dst + 2, v.z * w);
  unsafeAtomicAdd(dst + 3, v.w * w);
}

// ---------------- bias (+ optional ReLU) -------------------------------------
__global__ __launch_bounds__(256) void k_finalize(float* __restrict__ dst,
                                                  const float* __restrict__ agg,
                                                  const float* __restrict__ bias,
                                                  int total, int relu) {
  int i = blockIdx.x * blockDim.x + threadIdx.x;
  if (i < total) {
    float v = agg[i] + bias[i & 127];
    if (relu) v = fmaxf(v, 0.0f);
    dst[i] = v;
  }
}

// ---------------- readout: out[M,16] = H[M,128] * linW^T + linb (WMMA) -------
__global__ __launch_bounds__(32) void k_readout(const float* __restrict__ H,
                                                const float* __restrict__ linW,
                                                const float* __restrict__ linb,
                                                float* __restrict__ out, int M) {
  int lane = threadIdx.x & 31;
  int m0 = blockIdx.x * 16;
  int half = lane >> 4, l15 = lane & 15;
  v8f acc = {};
  const float* arow = H + (size_t)(m0 + l15) * NH;
  const float* brow = linW + (size_t)l15 * NH;   // B[k][n] = linW[n][k]
  #pragma unroll 4
  for (int k0 = 0; k0 < NH; k0 += 4) {
    int k = k0 + 2 * half;
    v2f a = { arow[k], arow[k + 1] };
    v2f b = { brow[k], brow[k + 1] };
    acc = __builtin_amdgcn_wmma_f32_16x16x4_f32(
        false, a, false, b, (short)0, acc, false, false);
  }
  float bb = linb[l15];
  int rbase = m0 + 8 * half;
  #pragma unroll
  for (int r = 0; r < 8; ++r)
    out[(size_t)(rbase + r) * NC + l15] = acc[r] + bb;
}

extern "C" void kernel_launch(void* const* d_in, const int* in_sizes, int n_in,
                              void* d_out, int out_size, void* d_ws, size_t ws_size,
                              hipStream_t stream) {
  const float* x    = (const float*)d_in[0];
  const int*   ei   = (const int*)  d_in[1];   // [2, NE]: row = targets, col = sources
  const float* W1   = (const float*)d_in[2];
  const float* b1   = (const float*)d_in[3];
  const float* W2   = (const float*)d_in[4];
  const float* b2   = (const float*)d_in[5];
  const float* linW = (const float*)d_in[6];   // [16,128]
  const float* linb = (const float*)d_in[7];
  float* out = (float*)d_out;

  const int* row = ei;
  const int* col = ei + NE;

  char* ws = (char*)d_ws;
  float* dinv = (float*)ws;                                 // NN floats (u32 cnt aliased)
  size_t off = ((size_t)NN * 4 + 4095) & ~(size_t)4095;
  float* bufA = (float*)(ws + off);                         // NN*128 floats
  float* bufB = (float*)(ws + off + (size_t)NN * NH * 4);   // NN*128 floats

  const int total = NN * NH;
  const dim3 eg((NE + 255) / 256), ng((NN + 255) / 256), tg((total + 255) / 256);
  const dim3 gemmGrid(NN / 16);            // 100000 / 16 == 6250 exactly
  const dim3 scatGrid((NE + 7) / 8);

  // normalization: deg -> dinv
  k_init_cnt<<<ng, 256, 0, stream>>>((unsigned int*)dinv, NN);
  k_count   <<<eg, 256, 0, stream>>>(row, (unsigned int*)dinv, NE);
  k_dinv    <<<ng, 256, 0, stream>>>(dinv, NN);

  // layer 1: hw1 = x@W1 (bufA); agg (bufB); h1 = relu(agg+b1) -> bufA
  k_gemm128 <<<gemmGrid, 256, 0, stream>>>(x, W1, bufA, NN);
  k_selfloop<<<tg, 256, 0, stream>>>(bufB, bufA, dinv, total);
  k_scatter <<<scatGrid, 256, 0, stream>>>(row, col, dinv, bufA, bufB, NE);
  k_finalize<<<tg, 256, 0, stream>>>(bufA, bufB, b1, total, 1);

  // layer 2: hw2 = h1@W2 (bufB); agg (bufA); h2 = agg+b2 -> bufB
  k_gemm128 <<<gemmGrid, 256, 0, stream>>>(bufA, W2, bufB, NN);
  k_selfloop<<<tg, 256, 0, stream>>>(bufA, bufB, dinv, total);
  k_scatter <<<scatGrid, 256, 0, stream>>>(row, col, dinv, bufB, bufA, NE);
  k_finalize<<<tg, 256, 0, stream>>>(bufB, bufA, b2, total, 0);

  // readout
  k_readout <<<gemmGrid, 32, 0, stream>>>(bufB, linW, linb, out, NN);
}